// HeteroMessagePassingAlongMP_45930380263451
// MI455X (gfx1250) — compile-verified
//
#include <hip/hip_runtime.h>
#include <cstdint>

#define THREADS 256
#define WAVE 32
#define WAVES_PER_BLOCK (THREADS / WAVE)
#define TILE 32        // edges staged per wave-tile
#define DIM 256        // feature dimension (fp32)
// per lane: 8 consecutive floats (2 x float4); 32 lanes cover one 256-float row

#if defined(__has_builtin)
#if __has_builtin(__builtin_amdgcn_global_load_async_to_lds_b32)
#define HAVE_ASYNC_LDS 1
#else
#define HAVE_ASYNC_LDS 0
#endif
#else
#define HAVE_ASYNC_LDS 0
#endif

__device__ __forceinline__ void wait_async_lds() {
#if defined(__has_builtin)
#if __has_builtin(__builtin_amdgcn_s_wait_asynccnt)
  __builtin_amdgcn_s_wait_asynccnt(0);
#else
  asm volatile("s_wait_asynccnt 0" ::: "memory");
#endif
#endif
  asm volatile("" ::: "memory");  // compiler barrier: force LDS re-read after wait
}

__global__ void zero_f32x4(float4* __restrict__ p, long long n4) {
  long long i = (long long)blockIdx.x * blockDim.x + threadIdx.x;
  long long stride = (long long)gridDim.x * blockDim.x;
  float4 z = make_float4(0.f, 0.f, 0.f, 0.f);
  for (; i < n4; i += stride) p[i] = z;
}

__global__ void relu_f32x4(float4* __restrict__ p, long long n4) {
  long long i = (long long)blockIdx.x * blockDim.x + threadIdx.x;
  long long stride = (long long)gridDim.x * blockDim.x;
  for (; i < n4; i += stride) {
    float4 v = p[i];
    v.x = fmaxf(v.x, 0.f);
    v.y = fmaxf(v.y, 0.f);
    v.z = fmaxf(v.z, 0.f);
    v.w = fmaxf(v.w, 0.f);
    p[i] = v;
  }
}

// One wave per 32-edge tile. Indices are staged into LDS with CDNA5 async
// global->LDS copies (ASYNCcnt), then each edge's 256-float source row is
// gathered with two global_load_b128 per lane and scatter-added with 8
// no-return global_atomic_add_f32 per lane into the accumulator. Row bases
// are forced scalar via readfirstlane so the gathers/atomics use the
// SGPR-base + per-lane-offset addressing form (address math on SALU).
template <bool RELU_GATHER>
__global__ void scatter_hop(const float* __restrict__ xsrc,
                            const int* __restrict__ esrc,
                            const int* __restrict__ edst,
                            float* __restrict__ acc, int n_edges) {
  __shared__ int lds_src[WAVES_PER_BLOCK][TILE];
  __shared__ int lds_dst[WAVES_PER_BLOCK][TILE];

  const int lane = threadIdx.x & (WAVE - 1);
  const int wid = threadIdx.x >> 5;
  const long long wave_gid = (long long)blockIdx.x * WAVES_PER_BLOCK + wid;
  const long long nwaves = (long long)gridDim.x * WAVES_PER_BLOCK;
  const int ntiles = (n_edges + TILE - 1) / TILE;
  const int col = lane * 8;  // this lane's 8 floats within a row

  for (long long t = wave_gid; t < ntiles; t += nwaves) {
    const int base = (int)t * TILE;
    int eidx = base + lane;
    if (eidx >= n_edges) eidx = n_edges - 1;  // clamped lanes are never consumed

#if HAVE_ASYNC_LDS
    __builtin_amdgcn_global_load_async_to_lds_b32(
        (int*)(esrc + eidx), (int*)&lds_src[wid][lane], 0, 0);
    __builtin_amdgcn_global_load_async_to_lds_b32(
        (int*)(edst + eidx), (int*)&lds_dst[wid][lane], 0, 0);
    wait_async_lds();
#else
    lds_src[wid][lane] = esrc[eidx];
    lds_dst[wid][lane] = edst[eidx];
#endif

    int cnt = n_edges - base;
    if (cnt > TILE) cnt = TILE;

    for (int j = 0; j < cnt; ++j) {
      // Wave-uniform indices: force into SGPRs for scalar address math.
      const int s = __builtin_amdgcn_readfirstlane(lds_src[wid][j]);
      const int d = __builtin_amdgcn_readfirstlane(lds_dst[wid][j]);
      const float4* rs = (const float4*)(xsrc + (long long)s * DIM + col);
      float* rd = acc + (long long)d * DIM + col;
      float4 a = rs[0];
      float4 b = rs[1];
      if (RELU_GATHER) {
        a.x = fmaxf(a.x, 0.f); a.y = fmaxf(a.y, 0.f);
        a.z = fmaxf(a.z, 0.f); a.w = fmaxf(a.w, 0.f);
        b.x = fmaxf(b.x, 0.f); b.y = fmaxf(b.y, 0.f);
        b.z = fmaxf(b.z, 0.f); b.w = fmaxf(b.w, 0.f);
      }
      unsafeAtomicAdd(rd + 0, a.x);
      unsafeAtomicAdd(rd + 1, a.y);
      unsafeAtomicAdd(rd + 2, a.z);
      unsafeAtomicAdd(rd + 3, a.w);
      unsafeAtomicAdd(rd + 4, b.x);
      unsafeAtomicAdd(rd + 5, b.y);
      unsafeAtomicAdd(rd + 6, b.z);
      unsafeAtomicAdd(rd + 7, b.w);
    }
  }
}

extern "C" void kernel_launch(void* const* d_in, const int* in_sizes, int n_in,
                              void* d_out, int out_size, void* d_ws,
                              size_t ws_size, hipStream_t stream) {
  const float* x_paper = (const float*)d_in[0];
  // d_in[1] (x_author) is dead per the reference: hop 1 overwrites author
  // features; only its row count matters.
  const int* e_pa_src = (const int*)d_in[2];
  const int* e_pa_dst = (const int*)d_in[3];
  const int* e_ap_src = (const int*)d_in[4];
  const int* e_ap_dst = (const int*)d_in[5];

  const long long n_paper = (long long)in_sizes[0] / DIM;
  const long long n_author = (long long)in_sizes[1] / DIM;
  const int e1 = in_sizes[2];
  const int e2 = in_sizes[4];

  float* x_a1 = (float*)d_ws;       // intermediate author accumulator
  float* out = (float*)d_out;       // final paper features

  const long long a4 = n_author * DIM / 4;
  const long long p4 = n_paper * DIM / 4;

  auto grid_for = [](long long n4) {
    long long b = (n4 + THREADS - 1) / THREADS;
    if (b > 8192) b = 8192;  // grid-stride handles the rest
    return (int)b;
  };

  // Re-zero accumulators every call (atomics accumulate; graph replays reuse buffers).
  zero_f32x4<<<grid_for(a4), THREADS, 0, stream>>>((float4*)x_a1, a4);
  zero_f32x4<<<grid_for(p4), THREADS, 0, stream>>>((float4*)out, p4);

  const int blocks1 = ((e1 + TILE - 1) / TILE + WAVES_PER_BLOCK - 1) / WAVES_PER_BLOCK;
  const int blocks2 = ((e2 + TILE - 1) / TILE + WAVES_PER_BLOCK - 1) / WAVES_PER_BLOCK;

  // Hop 1: paper -> author, plain gather (ReLU deferred to hop-2 read).
  scatter_hop<false><<<blocks1, THREADS, 0, stream>>>(x_paper, e_pa_src,
                                                      e_pa_dst, x_a1, e1);
  // Hop 2: relu(author) -> paper.
  scatter_hop<true><<<blocks2, THREADS, 0, stream>>>(x_a1, e_ap_src, e_ap_dst,
                                                     out, e2);
  // Final ReLU on the paper output.
  relu_f32x4<<<grid_for(p4), THREADS, 0, stream>>>((float4*)out, p4);
}